// GnnConv_88837103550598
// MI455X (gfx1250) — compile-verified
//
#include <hip/hip_runtime.h>

#define N_NODES 100000
#define N_EDGES 600000
#define EMB_DIM 128
#define ROWS_PER_BLOCK 80          // 5 sub-tiles of 16 rows; 100000 = 80 * 1250

typedef __attribute__((ext_vector_type(2))) float v2f;
typedef __attribute__((ext_vector_type(8))) float v8f;

// Hardware fp32 global atomic add (no return -> STOREcnt path), device scope.
__device__ __forceinline__ void gatomic_add_f32(float* addr, float v) {
    asm volatile("global_atomic_add_f32 %0, %1, off scope:SCOPE_DEV"
                 :: "v"(addr), "v"(v)
                 : "memory");
}

// ---------------------------------------------------------------- zero agg
__global__ __launch_bounds__(256) void zero_kernel(float4* __restrict__ p, int n4) {
    int i = blockIdx.x * 256 + threadIdx.x;
    if (i < n4) p[i] = make_float4(0.f, 0.f, 0.f, 0.f);
}

// --------------------------------------------- edge phase: aggHe[d] += Σ_f emb
// agg[n] = Σ_{dst=n} h[n]⊙he = h[n] ⊙ Σ he  -> h is never gathered here.
// One wave (32 lanes) per edge; lane l owns floats [4l, 4l+4).
__global__ __launch_bounds__(256) void edge_scatter_kernel(
    const int*   __restrict__ edge_feat,   // [E,3]
    const int*   __restrict__ dst,         // [E]
    const float* __restrict__ bond_emb,    // [3,8,128] (12 KB, cache-resident)
    float*       __restrict__ aggHe)       // [N,128] accumulator (in d_ws)
{
    const int e    = (int)((blockIdx.x * 256u + threadIdx.x) >> 5);
    const int lane = threadIdx.x & 31;
    if (e >= N_EDGES) return;

    const int d  = dst[e];
    const int f0 = edge_feat[e * 3 + 0];
    const int f1 = edge_feat[e * 3 + 1];
    const int f2 = edge_feat[e * 3 + 2];

    const int c = lane << 2;
    const float4 e0 = *(const float4*)(bond_emb + (0 * 8 + f0) * EMB_DIM + c);
    const float4 e1 = *(const float4*)(bond_emb + (1 * 8 + f1) * EMB_DIM + c);
    const float4 e2 = *(const float4*)(bond_emb + (2 * 8 + f2) * EMB_DIM + c);

    float* p = aggHe + (size_t)d * EMB_DIM + c;
    gatomic_add_f32(p + 0, e0.x + e1.x + e2.x);
    gatomic_add_f32(p + 1, e0.y + e1.y + e2.y);
    gatomic_add_f32(p + 2, e0.z + e1.z + e2.z);
    gatomic_add_f32(p + 3, e0.w + e1.w + e2.w);
}

// ------------------------------------- GEMM: out = (h ⊙ aggHe) @ W^T + b
// Block = 80 rows x 128 cols, 8 waves; wave w owns out cols [16w, 16w+16).
// B fragments (W rows for those 16 cols) are preloaded into 64 VGPRs once and
// reused across 5 row sub-tiles -> 5x less W refetch from L2.
// A tile (h ⊙ aggHe) staged once in LDS with pitch 132 (bank-conflict-free).
__global__ __launch_bounds__(256) void gemm_kernel(
    const float* __restrict__ h,        // [N,128]
    const float* __restrict__ aggHe,    // [N,128]
    const float* __restrict__ W,        // [128,128], B[k][n] = W[n][k]
    const float* __restrict__ bias,     // [128]
    float*       __restrict__ out)      // [N,128]
{
    __shared__ float As[ROWS_PER_BLOCK * 132];   // 42.24 KB

    const int row0 = blockIdx.x * ROWS_PER_BLOCK;
    const int tid  = threadIdx.x;

    // Stage A = h ⊙ aggHe for this 80-row tile (coalesced float4 loads).
    for (int i = tid; i < ROWS_PER_BLOCK * 32; i += 256) {   // 2560 float4s
        const int r = i >> 5;
        const int c = (i & 31) << 2;
        const size_t g = (size_t)(row0 + r) * EMB_DIM + c;
        const float4 hv = *(const float4*)(h + g);
        const float4 av = *(const float4*)(aggHe + g);
        *(float4*)(&As[r * 132 + c]) =
            make_float4(hv.x * av.x, hv.y * av.y, hv.z * av.z, hv.w * av.w);
    }
    __syncthreads();

    const int lane = tid & 31;
    const int wv   = tid >> 5;       // 0..7
    const int half = lane >> 4;      // 0: lanes 0-15, 1: lanes 16-31
    const int l16  = lane & 15;
    const int col0 = wv * 16;

    // Preload all 32 B fragments for this wave's 16 output columns.
    // ISA layout: B VGPR v holds K = 4j + 2*half + v (v = 0,1).
    const float* Wrow = W + (size_t)(col0 + l16) * EMB_DIM;   // B[k][n] = W[n][k]
    v2f bfrag[32];
#pragma unroll
    for (int j = 0; j < 32; ++j) {
        bfrag[j][0] = Wrow[4 * j + 2 * half + 0];
        bfrag[j][1] = Wrow[4 * j + 2 * half + 1];
    }

    const float bv = bias[col0 + l16];

    // 5 row sub-tiles of 16, reusing the register-resident B fragments.
    for (int rt = 0; rt < ROWS_PER_BLOCK / 16; ++rt) {
        const float* Arow = &As[(rt * 16 + l16) * 132];
        v8f acc = {};
#pragma unroll
        for (int j = 0; j < 32; ++j) {
            v2f a;
            a[0] = Arow[4 * j + 2 * half + 0];
            a[1] = Arow[4 * j + 2 * half + 1];
            acc = __builtin_amdgcn_wmma_f32_16x16x4_f32(
                /*neg_a=*/false, a, /*neg_b=*/false, bfrag[j],
                /*c_mod=*/(short)0, acc, /*reuse_a=*/false, /*reuse_b=*/false);
        }
        // C/D layout: VGPR r = row (r + 8*half), col = l16.
#pragma unroll
        for (int r = 0; r < 8; ++r) {
            out[(size_t)(row0 + rt * 16 + r + 8 * half) * EMB_DIM + col0 + l16] =
                acc[r] + bv;
        }
    }
}

// ----------------------------------------------------------------- launcher
extern "C" void kernel_launch(void* const* d_in, const int* in_sizes, int n_in,
                              void* d_out, int out_size, void* d_ws, size_t ws_size,
                              hipStream_t stream) {
    const float* h        = (const float*)d_in[0];
    const int*   edgef    = (const int*)d_in[1];
    const int*   dst      = (const int*)d_in[2];
    const float* bond_emb = (const float*)d_in[3];
    const float* W        = (const float*)d_in[4];
    const float* b        = (const float*)d_in[5];
    float*       out      = (float*)d_out;

    float* aggHe = (float*)d_ws;                 // N_NODES*EMB_DIM fp32 = 51.2 MB

    const int n4 = N_NODES * EMB_DIM / 4;
    zero_kernel<<<(n4 + 255) / 256, 256, 0, stream>>>((float4*)aggHe, n4);

    const int edge_blocks = (N_EDGES * 32 + 255) / 256;   // 1 wave per edge
    edge_scatter_kernel<<<edge_blocks, 256, 0, stream>>>(edgef, dst, bond_emb, aggHe);

    gemm_kernel<<<N_NODES / ROWS_PER_BLOCK, 256, 0, stream>>>(h, aggHe, W, b, out);
}